// QCNN_40321152974999
// MI455X (gfx1250) — compile-verified
//
#include <hip/hip_runtime.h>

#define NQ   10
#define DIM  1024
#define NL   6
#define BATCH_TOTAL 16384

typedef float v2f __attribute__((ext_vector_type(2)));
typedef float v8f __attribute__((ext_vector_type(8)));

__device__ __forceinline__ float2 cmul(float2 a, float2 b) {
  return make_float2(a.x * b.x - a.y * b.y, a.x * b.y + a.y * b.x);
}

// Composition of a layer's 10 CNOTs as an index permutation:
// final[i] = pre[g_0(g_1(...g_9(i)))], g_w(i) = i ^ (bit_c(i) << t_pos).
// This map is GF(2)-linear: perm(a ^ b) == perm(a) ^ perm(b), perm(0) == 0.
__device__ __forceinline__ int cnot_perm(int i, int r) {
  int idx = i;
#pragma unroll
  for (int w = 9; w >= 0; --w) {
    int cpos = 9 - w;
    int t = w + r; if (t >= NQ) t -= NQ;
    int tpos = 9 - t;
    idx ^= ((idx >> cpos) & 1) << tpos;
  }
  return idx;
}

__device__ __forceinline__ int maybe_perm(int v, int pr) {
  return pr ? cnot_perm(v, pr) : v;
}

// Disjoint-bit index contributions: logical index i = tcon ^ ncon ^ ycon.
// P0: i = (y<<6) | (t<<4) | nl         (y = bits 9:6)
// P1: i = (t<<8) | ((nl>>2)<<6) | (y<<2) | (nl&3)   (y = bits 5:2)
// P2: i = (t<<8) | (nl<<4) | y         (y = bits 3:0)
template <int P> __device__ __forceinline__ int ycon(int y) {
  return P == 0 ? (y << 6) : (P == 1 ? (y << 2) : y);
}
template <int P> __device__ __forceinline__ int ncon(int nl) {
  return P == 0 ? nl : (P == 1 ? (((nl >> 2) << 6) | (nl & 3)) : (nl << 4));
}
template <int P> __device__ __forceinline__ int tcon(int t) {
  return P == 0 ? (t << 4) : (t << 8);
}

// Build the wave's A-matrix fragments (16x4 f32 per K-chunk) for the 16x16
// complex gate of (layer l, pass p).  ISA layout: lane<16 -> M=lane, K={0,1};
// lane>=16 -> M=lane-16, K={2,3}; chunk kc adds 4*kc to K.
__device__ __forceinline__ void gen_frags(const float* __restrict__ qw, int l, int p,
                                          int lane, v2f* are, v2f* aim, v2f* ain) {
  float2 U[4][4];  // [position][2*xbit + ybit]
#pragma unroll
  for (int pos = 0; pos < 4; ++pos) {
    int q = 0; bool ident = false;
    if (p == 0) q = pos;
    else if (p == 1) q = 4 + pos;
    else { if (pos < 2) ident = true; else q = 8 + (pos - 2); }
    if (ident) {
      U[pos][0] = make_float2(1.f, 0.f); U[pos][1] = make_float2(0.f, 0.f);
      U[pos][2] = make_float2(0.f, 0.f); U[pos][3] = make_float2(1.f, 0.f);
    } else {
      const float* w = qw + ((l * NQ) + q) * 3;  // uniform -> scalar loads
      float phi = w[0], th = w[1], om = w[2];
      float st, ct;  __sincosf(0.5f * th, &st, &ct);
      float sa, ca;  __sincosf(0.5f * (phi + om), &sa, &ca);
      float sb, cb;  __sincosf(0.5f * (phi - om), &sb, &cb);
      U[pos][0] = make_float2(ct * ca, -ct * sa);   // u00 = e^{-i(phi+om)/2} c
      U[pos][1] = make_float2(-st * cb, -st * sb);  // u01 = -e^{+i(phi-om)/2} s
      U[pos][2] = make_float2(st * cb, -st * sb);   // u10 = e^{-i(phi-om)/2} s
      U[pos][3] = make_float2(ct * ca, ct * sa);    // u11 = e^{+i(phi+om)/2} c
    }
  }
  int x = lane & 15;
  int kbase = (lane >> 4) << 1;
#pragma unroll
  for (int kc = 0; kc < 4; ++kc) {
#pragma unroll
    for (int e = 0; e < 2; ++e) {
      int y = 4 * kc + kbase + e;
      float2 m = make_float2(1.f, 0.f);
#pragma unroll
      for (int pos = 0; pos < 4; ++pos) {
        int xb = (x >> (3 - pos)) & 1;
        int yb = (y >> (3 - pos)) & 1;
        m = cmul(m, U[pos][2 * xb + yb]);
      }
      if (e == 0) { are[kc].x = m.x; aim[kc].x = m.y; ain[kc].x = -m.y; }
      else        { are[kc].y = m.x; aim[kc].y = m.y; ain[kc].y = -m.y; }
    }
  }
}

// Apply one 16x16 complex gate along axis P to a 1024-amplitude state in LDS,
// in place, with V_WMMA_F32_16X16X4_F32.  pr != 0 fuses the previous layer's
// CNOT ring into the read addressing: since the permutation is GF(2)-linear
// and the index contributions are bit-disjoint, each read address is just
// perm(tcon) ^ perm(ncon) ^ perm(ycon), with the three parts hoisted.
template <int P>
__device__ __forceinline__ void apply_pass(float2* __restrict__ slot, int pr, int lane,
                                           const v2f* are, const v2f* aim,
                                           const v2f* ain) {
  int nl = lane & 15;
  int kb = (lane >> 4) << 1;
  int xb = (lane >> 4) << 3;

  int py[8];
#pragma unroll
  for (int kc = 0; kc < 4; ++kc)
#pragma unroll
    for (int e = 0; e < 2; ++e)
      py[2 * kc + e] = maybe_perm(ycon<P>(4 * kc + kb + e), pr);
  int rn = maybe_perm(ncon<P>(nl), pr);
  int wn = ncon<P>(nl);

#pragma unroll
  for (int t = 0; t < 4; ++t) {
    int rbase = maybe_perm(tcon<P>(t), pr) ^ rn;  // uniform perm -> scalar
    v2f bre[4], bim[4];
#pragma unroll
    for (int kc = 0; kc < 4; ++kc) {
      float2 c0 = slot[rbase ^ py[2 * kc]];
      float2 c1 = slot[rbase ^ py[2 * kc + 1]];
      v2f br; br.x = c0.x; br.y = c1.x;
      v2f bi; bi.x = c0.y; bi.y = c1.y;
      bre[kc] = br; bim[kc] = bi;
    }
    v8f dre  = {0.f, 0.f, 0.f, 0.f, 0.f, 0.f, 0.f, 0.f};
    v8f dim_ = {0.f, 0.f, 0.f, 0.f, 0.f, 0.f, 0.f, 0.f};
    // D_re = Re(A)Re(B) + (-Im(A))Im(B) ; D_im = Re(A)Im(B) + Im(A)Re(B)
#pragma unroll
    for (int kc = 0; kc < 4; ++kc)
      dre = __builtin_amdgcn_wmma_f32_16x16x4_f32(false, are[kc], false, bre[kc],
                                                  (short)0, dre, false, false);
#pragma unroll
    for (int kc = 0; kc < 4; ++kc)
      dre = __builtin_amdgcn_wmma_f32_16x16x4_f32(false, ain[kc], false, bim[kc],
                                                  (short)0, dre, false, false);
#pragma unroll
    for (int kc = 0; kc < 4; ++kc)
      dim_ = __builtin_amdgcn_wmma_f32_16x16x4_f32(false, are[kc], false, bim[kc],
                                                   (short)0, dim_, false, false);
#pragma unroll
    for (int kc = 0; kc < 4; ++kc)
      dim_ = __builtin_amdgcn_wmma_f32_16x16x4_f32(false, aim[kc], false, bre[kc],
                                                   (short)0, dim_, false, false);
    int wbase = tcon<P>(t) ^ wn;  // writes are never permuted
#pragma unroll
    for (int j = 0; j < 8; ++j)   // D vgpr j: rows j (lanes 0-15), j+8 (16-31)
      slot[wbase ^ ycon<P>(j + xb)] = make_float2(dre[j], dim_[j]);
  }
}

__global__ void __launch_bounds__(128)
qcnn_circuit_kernel(const float* __restrict__ x, const float* __restrict__ qw,
                    const float* __restrict__ fcw, const float* __restrict__ fcb,
                    float* __restrict__ out) {
  __shared__ float2 smem[8][DIM];  // 4 waves x 2 slots x 8KB = 64KB
  int tid = threadIdx.x;
  int wave = tid >> 5;
  int lane = tid & 31;
  int b0 = (int)blockIdx.x * 8 + wave * 2;

  __builtin_prefetch(&x[b0 * NQ], 0, 0);  // global_prefetch_b8

  // ---- build initial product states: amp[i] = (prod_q bit? s_q : c_q) * (-i)^popc(i)
#pragma unroll 1
  for (int s = 0; s < 2; ++s) {
    int b = b0 + s;
    float2* slot = smem[wave * 2 + s];
    float cq[NQ], sq[NQ];
#pragma unroll
    for (int q = 0; q < NQ; ++q) {
      float xv = x[b * NQ + q];
      __sincosf(0.5f * xv, &sq[q], &cq[q]);
    }
#pragma unroll 1
    for (int j = 0; j < 32; ++j) {
      int i = lane * 32 + j;
      float m = 1.f;
#pragma unroll
      for (int q = 0; q < NQ; ++q)
        m *= ((i >> (9 - q)) & 1) ? sq[q] : cq[q];
      int k = __popc(i) & 3;
      float2 amp;
      if (k == 0)      amp = make_float2(m, 0.f);
      else if (k == 1) amp = make_float2(0.f, -m);
      else if (k == 2) amp = make_float2(-m, 0.f);
      else             amp = make_float2(0.f, m);
      slot[i] = amp;
    }
  }

  // ---- 6 layers; CNOT ring of layer l-1 fused into pass-0 reads of layer l
#pragma unroll 1
  for (int l = 0; l < NL; ++l) {
    int rprev = (l == 0) ? 0 : ((l - 1) % (NQ - 1)) + 1;
    {
      v2f are[4], aim[4], ain[4];
      gen_frags(qw, l, 0, lane, are, aim, ain);
#pragma unroll 1
      for (int s = 0; s < 2; ++s)
        apply_pass<0>(smem[wave * 2 + s], rprev, lane, are, aim, ain);
    }
    {
      v2f are[4], aim[4], ain[4];
      gen_frags(qw, l, 1, lane, are, aim, ain);
#pragma unroll 1
      for (int s = 0; s < 2; ++s)
        apply_pass<1>(smem[wave * 2 + s], 0, lane, are, aim, ain);
    }
    {
      v2f are[4], aim[4], ain[4];
      gen_frags(qw, l, 2, lane, are, aim, ain);
#pragma unroll 1
      for (int s = 0; s < 2; ++s)
        apply_pass<2>(smem[wave * 2 + s], 0, lane, are, aim, ain);
    }
  }

  // ---- measurement: out[b] = sum_i |amp|^2 * (sum_n w_n * sign_n(i)) + fc_b,
  // reading through the final layer's CNOT permutation (split linearly:
  // perm(i) = perm(lane<<5) ^ perm(j), the latter uniform -> scalar).
  int rlast = ((NL - 1) % (NQ - 1)) + 1;
  float wv[NQ]; float base = 0.f;
#pragma unroll
  for (int n = 0; n < NQ; ++n) { wv[n] = fcw[n]; base += wv[n]; }
  float fb = fcb[0];
  int pl = cnot_perm(lane << 5, rlast);
#pragma unroll 1
  for (int s = 0; s < 2; ++s) {
    int b = b0 + s;
    float2* slot = smem[wave * 2 + s];
    float acc = 0.f;
#pragma unroll 1
    for (int j = 0; j < 32; ++j) {
      int i = lane * 32 + j;
      int ri = pl ^ cnot_perm(j, rlast);  // uniform part scalarizes
      float2 a = slot[ri];
      float pm = a.x * a.x + a.y * a.y;
      float coef = base;
#pragma unroll
      for (int n = 0; n < NQ; ++n)
        coef -= ((i >> (9 - n)) & 1) ? 2.f * wv[n] : 0.f;
      acc += pm * coef;
    }
#pragma unroll
    for (int off = 16; off > 0; off >>= 1)
      acc += __shfl_xor(acc, off, 32);
    if (lane == 0) out[b] = acc + fb;
  }
}

extern "C" void kernel_launch(void* const* d_in, const int* in_sizes, int n_in,
                              void* d_out, int out_size, void* d_ws, size_t ws_size,
                              hipStream_t stream) {
  (void)in_sizes; (void)n_in; (void)out_size; (void)d_ws; (void)ws_size;
  const float* x   = (const float*)d_in[0];  // (16384, 10) f32
  const float* qw  = (const float*)d_in[1];  // (6, 10, 3)  f32
  const float* fcw = (const float*)d_in[2];  // (1, 10)     f32
  const float* fcb = (const float*)d_in[3];  // (1,)        f32
  float* out = (float*)d_out;                // (16384, 1)  f32

  dim3 grid(BATCH_TOTAL / 8);  // 4 waves/block x 2 states/wave
  dim3 block(128);
  hipLaunchKernelGGL(qcnn_circuit_kernel, grid, block, 0, stream,
                     x, qw, fcw, fcb, out);
}